// GauntProductBasisWithWeight_74723841016343
// MI455X (gfx1250) — compile-verified
//
#include <hip/hip_runtime.h>

typedef __attribute__((ext_vector_type(2))) float v2f;
typedef __attribute__((ext_vector_type(8))) float v8f;

// static problem sizes
#define NCH      262144     // 4096 atoms * 64 mul
#define W_STRIDE 192        // w: (3, 64, 3)

// workspace layout (floats)
#define OFF_CP 0            // CP[d1*9+d2][p*9+q][re,im]  : 81*81*2 = 13122
#define OFF_Z3 13122        // Z3[do*9+d3][a*9+b][re,im]  : 81*81*2 = 13122
#define OFF_B  26244        // B fragments               : 21*6*32*2 = 8064
#define OFF_C1 34308        // C1[do*9+d1]               : 81

// ---------------------------------------------------------------------------
// setup 1: pairwise convolutions of y, and y*z correlations
//   CP[d1,d2](p,q) = sum_{u,v} y[d1](u,v) * y[d2](p-u,q-v)      (9x9 complex)
//   Z3[do,d3](a,b) = sum_{r,s} y[d3](r,s) * z[do](a+r,b+s)      (9x9 complex)
// ---------------------------------------------------------------------------
__global__ void gnt_setup1(const float* __restrict__ yre, const float* __restrict__ yim,
                           const float* __restrict__ zre, const float* __restrict__ zim,
                           float* __restrict__ ws) {
  int t = threadIdx.x;
  if (t < 81) {
    int d1 = t / 9, d2 = t % 9;
    for (int p = 0; p < 9; ++p)
      for (int q = 0; q < 9; ++q) {
        float re = 0.f, im = 0.f;
        int u0 = (p > 4) ? p - 4 : 0, u1 = (p < 4) ? p : 4;
        int v0 = (q > 4) ? q - 4 : 0, v1 = (q < 4) ? q : 4;
        for (int u = u0; u <= u1; ++u)
          for (int v = v0; v <= v1; ++v) {
            float ar = yre[d1 * 25 + u * 5 + v], ai = yim[d1 * 25 + u * 5 + v];
            float br = yre[d2 * 25 + (p - u) * 5 + (q - v)];
            float bi = yim[d2 * 25 + (p - u) * 5 + (q - v)];
            re += ar * br - ai * bi;
            im += ar * bi + ai * br;
          }
        ws[OFF_CP + (t * 81 + p * 9 + q) * 2 + 0] = re;
        ws[OFF_CP + (t * 81 + p * 9 + q) * 2 + 1] = im;
      }
  } else if (t < 162) {
    int u = t - 81;
    int dO = u / 9, d3 = u % 9;
    for (int a = 0; a < 9; ++a)
      for (int b = 0; b < 9; ++b) {
        float re = 0.f, im = 0.f;
        for (int r = 0; r < 5; ++r)
          for (int s = 0; s < 5; ++s) {
            float ar = yre[d3 * 25 + r * 5 + s], ai = yim[d3 * 25 + r * 5 + s];
            float br = zre[dO * 169 + (a + r) * 13 + (b + s)];
            float bi = zim[dO * 169 + (a + r) * 13 + (b + s)];
            re += ar * br - ai * bi;
            im += ar * bi + ai * br;
          }
        ws[OFF_Z3 + (u * 81 + a * 9 + b) * 2 + 0] = re;
        ws[OFF_Z3 + (u * 81 + a * 9 + b) * 2 + 1] = im;
      }
  }
}

// ---------------------------------------------------------------------------
// setup 2: build constant GEMM B matrix (84 x 96, fragment layout) and C1
//   cols 0..80 : C3[do,d1,d2,d3]/169  at col = d3*9+do, row K = d1*9+d2
//   cols 81..89: C2[do,d1,d2]/13      at col = 81+do
//   rows 81..83, cols 90..95: zero padding
// fragment layout: frag index e = ((k*6+nt)*32+lane)*2+j holds
//   B[K = 4k + 2*(lane/16) + j][N = nt*16 + lane%16]
// ---------------------------------------------------------------------------
__global__ void gnt_setup2(const float* __restrict__ yre, const float* __restrict__ yim,
                           const float* __restrict__ zre, const float* __restrict__ zim,
                           float* __restrict__ ws) {
  int gid = blockIdx.x * blockDim.x + threadIdx.x;
  if (gid < 8064) {
    int j = gid & 1;
    int t2 = gid >> 1;
    int lane = t2 & 31;
    int t3 = t2 >> 5;
    int nt = t3 % 6;
    int k = t3 / 6;
    int K = 4 * k + 2 * (lane >> 4) + j;
    int Nc = nt * 16 + (lane & 15);
    float val = 0.f;
    if (K < 81 && Nc < 90) {
      int cp_base = OFF_CP + K * 81 * 2;
      if (Nc < 81) {
        int d3 = Nc / 9, dO = Nc % 9;
        int z3_base = OFF_Z3 + (dO * 9 + d3) * 81 * 2;
        float s = 0.f;
        for (int g = 0; g < 81; ++g) {
          float cr = ws[cp_base + g * 2], ci = ws[cp_base + g * 2 + 1];
          float zr = ws[z3_base + g * 2], zi = ws[z3_base + g * 2 + 1];
          s += cr * zr - ci * zi;
        }
        val = s * (1.0f / 169.0f);
      } else {
        int dO = Nc - 81;
        float s = 0.f;
        for (int p = 0; p < 9; ++p)
          for (int q = 0; q < 9; ++q) {
            float cr = ws[cp_base + (p * 9 + q) * 2], ci = ws[cp_base + (p * 9 + q) * 2 + 1];
            float zr = zre[dO * 169 + (p + 2) * 13 + (q + 2)];
            float zi = zim[dO * 169 + (p + 2) * 13 + (q + 2)];
            s += cr * zr - ci * zi;
          }
        val = s * (1.0f / 13.0f);
      }
    }
    ws[OFF_B + gid] = val;
  } else if (gid < 8064 + 81) {
    int d = gid - 8064;
    int dO = d / 9, d1 = d % 9;
    float s = 0.f;
    for (int u = 0; u < 5; ++u)
      for (int v = 0; v < 5; ++v) {
        float ar = yre[d1 * 25 + u * 5 + v], ai = yim[d1 * 25 + u * 5 + v];
        float zr = zre[dO * 169 + (u + 4) * 13 + (v + 4)];
        float zi = zim[dO * 169 + (u + 4) * 13 + (v + 4)];
        s += ar * zr - ai * zi;
      }
    ws[OFF_C1 + d] = s;
  }
}

// ---------------------------------------------------------------------------
// main fused kernel: 256 threads = 8 waves; each wave owns 16 channels.
// LDS (dynamic): Bfrag 8064 f | per wave: xw0 144 | xw1 144 | xw2 144 | D 1600
// ---------------------------------------------------------------------------
__global__ __launch_bounds__(256)
void gnt_main(const float* __restrict__ x, const float* __restrict__ w,
              const float* __restrict__ ws, const float* __restrict__ path_w,
              float* __restrict__ out) {
  extern __shared__ float lds[];
  const int tid = threadIdx.x;
  const int wv = tid >> 5;
  const int lane = tid & 31;
  const int h = lane >> 4;     // half-wave index (K split)
  const int r = lane & 15;     // row within the 16-row tile

  float* Bf   = lds;                        // 8064 floats (shared by WG)
  float* wbs  = lds + 8064 + wv * 2032;     // per-wave region
  float* xw0s = wbs;                        // 16*9
  float* xw1s = wbs + 144;                  // 16*9
  float* xw2s = wbs + 288;                  // 16*9
  float* Dbuf = wbs + 432;                  // 16*100 (stride 100: bank-clean)

  // cooperative load of constant B fragments
  for (int i = tid; i < 8064; i += 256) Bf[i] = ws[OFF_B + i];

  const int rowbase = (blockIdx.x * 8 + wv) * 16;

  // phase 1: per-channel weighted inputs xw_v[d] = x[d] * w[v][m][deg(d)]
  if (lane < 16) {
    int ch = rowbase + lane;
    int m = ch & 63;
    float w0[3], w1[3], w2[3];
    #pragma unroll
    for (int l = 0; l < 3; ++l) {
      w0[l] = w[0 * W_STRIDE + m * 3 + l];
      w1[l] = w[1 * W_STRIDE + m * 3 + l];
      w2[l] = w[2 * W_STRIDE + m * 3 + l];
    }
    #pragma unroll
    for (int d = 0; d < 9; ++d) {
      int l = (d == 0) ? 0 : ((d < 4) ? 1 : 2);
      float xv = x[ch * 9 + d];
      xw0s[lane * 9 + d] = xv * w0[l];
      xw1s[lane * 9 + d] = xv * w1[l];
      xw2s[lane * 9 + d] = xv * w2[l];
    }
  }
  __syncthreads();

  // phase 2: A fragments (P2 outer product) directly in the 16x4 f32 layout:
  // lane holds A[row=r][K = 4k + 2h + j], K = d1*9+d2, value xw0[d1]*xw1[d2]
  float a0[21], a1[21];
  #pragma unroll
  for (int k = 0; k < 21; ++k) {
    int K0 = 4 * k + 2 * h;
    int K1 = K0 + 1;
    a0[k] = (K0 < 81) ? xw0s[r * 9 + K0 / 9] * xw1s[r * 9 + K0 % 9] : 0.f;
    a1[k] = (K1 < 81) ? xw0s[r * 9 + K1 / 9] * xw1s[r * 9 + K1 % 9] : 0.f;
  }

  // phase 3: GEMM (16 x 84) @ (84 x 96) via V_WMMA_F32_16X16X4_F32
  v8f acc[6];
  #pragma unroll
  for (int nt = 0; nt < 6; ++nt)
    #pragma unroll
    for (int e = 0; e < 8; ++e) acc[nt][e] = 0.f;

  #pragma unroll
  for (int k = 0; k < 21; ++k) {
    v2f a;
    a.x = a0[k];
    a.y = a1[k];
    #pragma unroll
    for (int nt = 0; nt < 6; ++nt) {
      int base = ((k * 6 + nt) * 32 + lane) * 2;
      v2f b;
      b.x = Bf[base];
      b.y = Bf[base + 1];
      acc[nt] = __builtin_amdgcn_wmma_f32_16x16x4_f32(
          /*neg_a=*/false, a, /*neg_b=*/false, b,
          /*c_mod=*/(short)0, acc[nt], /*reuse_a=*/false, /*reuse_b=*/false);
    }
  }

  // phase 4: stage D (C/D layout: N = lane%16, M = vgpr + 8*(lane/16))
  #pragma unroll
  for (int nt = 0; nt < 6; ++nt)
    #pragma unroll
    for (int rr = 0; rr < 8; ++rr)
      Dbuf[(rr + 8 * h) * 100 + nt * 16 + r] = acc[nt][rr];

  __syncthreads();

  // phase 5: epilogue  out[do] = (C2term + sum_d3 xw2[d3]*Q[d3,do] + linear) * path_w
  {
    int ch = rowbase + r;
    float xw2l[9], xw0l[9];
    #pragma unroll
    for (int d = 0; d < 9; ++d) {
      xw2l[d] = xw2s[r * 9 + d];
      xw0l[d] = xw0s[r * 9 + d];
    }
    for (int dO = h; dO < 9; dO += 2) {
      float s = Dbuf[r * 100 + 81 + dO];            // quadratic term
      #pragma unroll
      for (int d3 = 0; d3 < 9; ++d3)                // cubic term
        s += xw2l[d3] * Dbuf[r * 100 + 9 * d3 + dO];
      #pragma unroll
      for (int d1 = 0; d1 < 9; ++d1)                // linear term
        s += ws[OFF_C1 + dO * 9 + d1] * xw0l[d1];
      out[ch * 9 + dO] = s * path_w[dO];
    }
  }
}

// ---------------------------------------------------------------------------
extern "C" void kernel_launch(void* const* d_in, const int* in_sizes, int n_in,
                              void* d_out, int out_size, void* d_ws, size_t ws_size,
                              hipStream_t stream) {
  const float* x      = (const float*)d_in[0]; // (4096, 64, 9)
  const float* w      = (const float*)d_in[1]; // (3, 64, 3)
  const float* yre    = (const float*)d_in[2]; // (9, 5, 5)
  const float* yim    = (const float*)d_in[3];
  const float* zre    = (const float*)d_in[4]; // (9, 13, 13)
  const float* zim    = (const float*)d_in[5];
  // d_in[6] = a_w (degree-alignment structure is hardcoded: l(d) = 0,1,1,1,2..2)
  const float* path_w = (const float*)d_in[7]; // (9,)
  float* ws  = (float*)d_ws;
  float* out = (float*)d_out;

  gnt_setup1<<<1, 192, 0, stream>>>(yre, yim, zre, zim, ws);
  gnt_setup2<<<32, 256, 0, stream>>>(yre, yim, zre, zim, ws);

  const int shmem_bytes = (8064 + 8 * 2032) * 4;  // 97280 B dynamic LDS
  gnt_main<<<NCH / 128, 256, shmem_bytes, stream>>>(x, w, ws, path_w, out);
}